// GIN_22857815949371
// MI455X (gfx1250) — compile-verified
//
#include <hip/hip_runtime.h>

typedef float v2f __attribute__((ext_vector_type(2)));
typedef float v8f __attribute__((ext_vector_type(8)));

#define N_NODES 20000
#define F_IN    256
#define H_DIM   512
#define BN_EPS  1e-5f

// ---------------------------------------------------------------------------
// fp32 WMMA: D = A(16x4) * B(4x16) + C, full fp32 precision on matrix units.
// ---------------------------------------------------------------------------
__device__ __forceinline__ v8f wmma_f32(v2f a, v2f b, v8f c) {
    return __builtin_amdgcn_wmma_f32_16x16x4_f32(
        /*neg_a=*/false, a, /*neg_b=*/false, b,
        /*c_mod=*/(short)0, c, /*reuse_a=*/false, /*reuse_b=*/false);
}

// ---------------------------------------------------------------------------
// hin = (1 + eps) * x   (vectorized float4; fully initializes the scatter
// accumulator so no memset is needed)
// ---------------------------------------------------------------------------
__global__ __launch_bounds__(256) void init_self(const float* __restrict__ x,
                                                 const float* __restrict__ epsPtr,
                                                 float* __restrict__ out,
                                                 int total4) {
    int i = blockIdx.x * blockDim.x + threadIdx.x;
    if (i >= total4) return;
    float s = 1.0f + epsPtr[0];
    float4 v = reinterpret_cast<const float4*>(x)[i];
    v.x *= s; v.y *= s; v.z *= s; v.w *= s;
    reinterpret_cast<float4*>(out)[i] = v;
}

// ---------------------------------------------------------------------------
// scatter-add: out[dst[e]] += x[src[e]]  (per edge, 4 features per thread)
// ---------------------------------------------------------------------------
__global__ __launch_bounds__(256) void scatter_add(const float* __restrict__ x,
                                                   const int* __restrict__ srcIdx,
                                                   const int* __restrict__ dstIdx,
                                                   float* __restrict__ out,
                                                   int nEdges, int F) {
    int chunks = F >> 2;
    int gid = blockIdx.x * blockDim.x + threadIdx.x;
    if (gid >= nEdges * chunks) return;
    int e = gid / chunks;
    int c = (gid - e * chunks) << 2;
    int s = srcIdx[e];
    int d = dstIdx[e];
    float4 v = *reinterpret_cast<const float4*>(&x[s * F + c]);
    float* o = &out[d * F + c];
    atomicAdd(o + 0, v.x);
    atomicAdd(o + 1, v.y);
    atomicAdd(o + 2, v.z);
    atomicAdd(o + 3, v.w);
}

// ---------------------------------------------------------------------------
// Column-wise batch-norm statistics over t[N, 512]:
//   scale = g * rsqrt(var + eps), shift = be - mu * scale
// One block per 64 columns; 4 row-groups of 64 lanes each.
// ---------------------------------------------------------------------------
__global__ __launch_bounds__(256) void bn_stats(const float* __restrict__ t,
                                                const float* __restrict__ g,
                                                const float* __restrict__ be,
                                                float* __restrict__ scaleOut,
                                                float* __restrict__ shiftOut) {
    int lane = threadIdx.x & 63;
    int rg   = threadIdx.x >> 6;
    int col  = blockIdx.x * 64 + lane;
    float s = 0.0f, q = 0.0f;
    for (int r = rg; r < N_NODES; r += 4) {
        float v = t[r * H_DIM + col];
        s += v;
        q += v * v;
    }
    __shared__ float ssum[4][64];
    __shared__ float ssq[4][64];
    ssum[rg][lane] = s;
    ssq[rg][lane]  = q;
    __syncthreads();
    if (threadIdx.x < 64) {
        float S = 0.0f, Q = 0.0f;
#pragma unroll
        for (int i = 0; i < 4; ++i) { S += ssum[i][threadIdx.x]; Q += ssq[i][threadIdx.x]; }
        int c = blockIdx.x * 64 + threadIdx.x;
        float invN = 1.0f / (float)N_NODES;
        float mu  = S * invN;
        float var = Q * invN - mu * mu;
        float sc  = g[c] * rsqrtf(var + BN_EPS);
        scaleOut[c] = sc;
        shiftOut[c] = be[c] - mu * sc;
    }
}

// ---------------------------------------------------------------------------
// Tiled fp32 WMMA GEMM: out[nRows, H_DIM] = f(A[nRows, K]) @ W[K, H_DIM] + bias
// Block tile 128(M) x 64(N), K-step 16. 8 waves: 4(M) x 2(N), each wave a
// 32x32 tile (2x2 WMMA accumulators).
//
// A tile: staged manually (handles row guard + optional fused BN+ReLU),
//   row-major in LDS with stride 20 floats (80 B): A fragments are single
//   conflict-free ds_load_b64 (20*dl == 2 mod 64 has no solution, so the two
//   half-wave K-offsets never collide).
// B tile: pure in-bounds 2D copy -> staged with GLOBAL_LOAD_ASYNC_TO_LDS_B128
//   (ASYNCcnt), issued before A staging so the DMA overlaps the VALU work,
//   completed with s_wait_asynccnt before the barrier.
// ---------------------------------------------------------------------------
__global__ __launch_bounds__(256) void gemm_wmma(const float* __restrict__ A,
                                                 const float* __restrict__ W,
                                                 const float* __restrict__ bias,
                                                 float* __restrict__ out,
                                                 int nRows, int K,
                                                 const float* __restrict__ aScale,
                                                 const float* __restrict__ aShift,
                                                 int epiRelu) {
    __shared__ __align__(16) float As[128][20];  // [row][k], padded stride 20
    __shared__ __align__(16) float Bs[16][72];   // [k][col], padded stride 72

    const int tid  = threadIdx.x;
    const int lane = tid & 31;
    const int wave = tid >> 5;
    const int wm   = wave >> 1;          // 0..3 -> 32-row slice
    const int wn   = wave & 1;           // 0..1 -> 32-col slice

    const int row0 = blockIdx.y * 128;   // block row base
    const int col0 = blockIdx.x * 64;    // block col base

    v8f acc[2][2];
#pragma unroll
    for (int i = 0; i < 2; ++i)
#pragma unroll
        for (int j = 0; j < 2; ++j)
            acc[i][j] = (v8f){0.f, 0.f, 0.f, 0.f, 0.f, 0.f, 0.f, 0.f};

    const int half  = lane >> 4;         // 0: K pair {0,1}, 1: K pair {2,3}
    const int lid16 = lane & 15;

    // per-thread B-copy coordinates (constant across K loop)
    const int bkr = tid >> 4;            // 0..15 (k row)
    const int bnc = (tid & 15) << 2;     // 0..60 (col, float4)
    const unsigned bLds = (unsigned)(size_t)&Bs[bkr][bnc];

    for (int k0 = 0; k0 < K; k0 += 16) {
        // ---- B tile: async DMA global -> LDS (16 B per thread, 4 KB tile)
        {
            unsigned long long ga =
                (unsigned long long)(size_t)&W[(k0 + bkr) * H_DIM + col0 + bnc];
            asm volatile("global_load_async_to_lds_b128 %0, %1, off"
                         :: "v"(bLds), "v"(ga)
                         : "memory");
        }

        // ---- A tile (128 rows x 16 k), row-major, optional fused BN+ReLU
#pragma unroll
        for (int it = 0; it < 2; ++it) {
            int L = tid + it * 256;            // 0..511 float4 slots
            int rowIdx = L >> 2;               // 0..127
            int kc = (L & 3) << 2;             // 0,4,8,12
            int arow = row0 + rowIdx;
            float4 v = make_float4(0.f, 0.f, 0.f, 0.f);
            if (arow < nRows)
                v = *reinterpret_cast<const float4*>(&A[arow * K + k0 + kc]);
            if (aScale) {
                float s0 = aScale[k0 + kc + 0], h0 = aShift[k0 + kc + 0];
                float s1 = aScale[k0 + kc + 1], h1 = aShift[k0 + kc + 1];
                float s2 = aScale[k0 + kc + 2], h2 = aShift[k0 + kc + 2];
                float s3 = aScale[k0 + kc + 3], h3 = aShift[k0 + kc + 3];
                v.x = fmaxf(fmaf(v.x, s0, h0), 0.f);
                v.y = fmaxf(fmaf(v.y, s1, h1), 0.f);
                v.z = fmaxf(fmaf(v.z, s2, h2), 0.f);
                v.w = fmaxf(fmaf(v.w, s3, h3), 0.f);
            }
            *reinterpret_cast<float4*>(&As[rowIdx][kc]) = v;
        }

        // async B copy must have landed before anyone reads Bs
        asm volatile("s_wait_asynccnt 0x0" ::: "memory");
        __syncthreads();

        // ---- compute: 4 K-substeps of 4, 2x2 WMMA tiles per wave
#pragma unroll
        for (int k4 = 0; k4 < 16; k4 += 4) {
            int kk = k4 + (half << 1);
            int ar = wm * 32 + lid16;
            v2f a0 = *reinterpret_cast<const v2f*>(&As[ar][kk]);
            v2f a1 = *reinterpret_cast<const v2f*>(&As[ar + 16][kk]);
            v2f b0, b1;
            b0.x = Bs[kk][wn * 32 + lid16];
            b0.y = Bs[kk + 1][wn * 32 + lid16];
            b1.x = Bs[kk][wn * 32 + 16 + lid16];
            b1.y = Bs[kk + 1][wn * 32 + 16 + lid16];
            acc[0][0] = wmma_f32(a0, b0, acc[0][0]);
            acc[0][1] = wmma_f32(a0, b1, acc[0][1]);
            acc[1][0] = wmma_f32(a1, b0, acc[1][0]);
            acc[1][1] = wmma_f32(a1, b1, acc[1][1]);
        }
        __syncthreads();
    }

    // ---- epilogue: bias (+ optional ReLU), guarded stores
#pragma unroll
    for (int tm = 0; tm < 2; ++tm) {
#pragma unroll
        for (int tn = 0; tn < 2; ++tn) {
            int colBase = col0 + wn * 32 + tn * 16 + lid16;
            float bv = bias[colBase];
#pragma unroll
            for (int r = 0; r < 8; ++r) {
                int row = row0 + wm * 32 + tm * 16 + r + (half << 3);
                float v = acc[tm][tn][r] + bv;
                if (epiRelu) v = fmaxf(v, 0.f);
                if (row < nRows) out[row * H_DIM + colBase] = v;
            }
        }
    }
}

// ---------------------------------------------------------------------------
// Host-side orchestration (graph-capture safe: only kernel launches on stream)
// ---------------------------------------------------------------------------
extern "C" void kernel_launch(void* const* d_in, const int* in_sizes, int n_in,
                              void* d_out, int out_size, void* d_ws, size_t ws_size,
                              hipStream_t stream) {
    const float* x    = (const float*)d_in[0];
    const int*   edge = (const int*)d_in[1];
    const float* eps1 = (const float*)d_in[2];
    const float* W1   = (const float*)d_in[3];
    const float* b1   = (const float*)d_in[4];
    const float* g1   = (const float*)d_in[5];
    const float* be1  = (const float*)d_in[6];
    const float* W2   = (const float*)d_in[7];
    const float* b2   = (const float*)d_in[8];
    const float* eps2 = (const float*)d_in[9];
    const float* W3   = (const float*)d_in[10];
    const float* b3   = (const float*)d_in[11];
    const float* g2   = (const float*)d_in[12];
    const float* be2  = (const float*)d_in[13];
    const float* W4   = (const float*)d_in[14];
    const float* b4   = (const float*)d_in[15];

    const int E = in_sizes[1] / 2;
    const int* srcIdx = edge;
    const int* dstIdx = edge + E;

    // workspace layout (floats): hin1 | bufA | bufB | sc1 sh1 sc2 sh2
    float* ws   = (float*)d_ws;
    float* hin1 = ws;                                  // N x 256
    float* bufA = hin1 + (size_t)N_NODES * F_IN;       // N x 512 (t1, then hin2)
    float* bufB = bufA + (size_t)N_NODES * H_DIM;      // N x 512 (h1, then t2)
    float* sc1  = bufB + (size_t)N_NODES * H_DIM;
    float* sh1  = sc1 + H_DIM;
    float* sc2  = sh1 + H_DIM;
    float* sh2  = sc2 + H_DIM;

    dim3 blk(256);
    dim3 gemmGrid(H_DIM / 64, (N_NODES + 127) / 128);

    // ---------------- Layer 1 ----------------
    {
        int total4 = N_NODES * F_IN / 4;
        init_self<<<(total4 + 255) / 256, blk, 0, stream>>>(x, eps1, hin1, total4);
        int nThreads = E * (F_IN / 4);
        scatter_add<<<(nThreads + 255) / 256, blk, 0, stream>>>(x, srcIdx, dstIdx, hin1, E, F_IN);
        // t1 = hin1 @ W1 + b1
        gemm_wmma<<<gemmGrid, blk, 0, stream>>>(hin1, W1, b1, bufA, N_NODES, F_IN,
                                                nullptr, nullptr, 0);
        bn_stats<<<H_DIM / 64, blk, 0, stream>>>(bufA, g1, be1, sc1, sh1);
        // h1 = relu( relu(bn(t1)) @ W2 + b2 )
        gemm_wmma<<<gemmGrid, blk, 0, stream>>>(bufA, W2, b2, bufB, N_NODES, H_DIM,
                                                sc1, sh1, 1);
    }

    // ---------------- Layer 2 ----------------
    {
        int total4 = N_NODES * H_DIM / 4;
        init_self<<<(total4 + 255) / 256, blk, 0, stream>>>(bufB, eps2, bufA, total4);
        int nThreads = E * (H_DIM / 4);
        scatter_add<<<(nThreads + 255) / 256, blk, 0, stream>>>(bufB, srcIdx, dstIdx, bufA, E, H_DIM);
        // t2 = hin2 @ W3 + b3   (bufA -> bufB; h1 dead)
        gemm_wmma<<<gemmGrid, blk, 0, stream>>>(bufA, W3, b3, bufB, N_NODES, H_DIM,
                                                nullptr, nullptr, 0);
        bn_stats<<<H_DIM / 64, blk, 0, stream>>>(bufB, g2, be2, sc2, sh2);
        // out = relu( relu(bn(t2)) @ W4 + b4 )
        gemm_wmma<<<gemmGrid, blk, 0, stream>>>(bufB, W4, b4, (float*)d_out, N_NODES, H_DIM,
                                                sc2, sh2, 1);
    }
}